// SpectralPooling_19585050870114
// MI455X (gfx1250) — compile-verified
//
#include <hip/hip_runtime.h>
#include <hip/hip_bf16.h>

// ---------------------------------------------------------------------------
// SpectralPooling on MI455X (gfx1250)
//   out[b,c,k,j,i] = sum_{d,h,w} M[k,d] M[j,h] M[i,w] x[b,c,d,h,w]
//   with M = C32^T @ C64[:32,:]  (32x64, identical for all three axes)
//
// One workgroup (32 wave32s) per (b,c) slab. x read exactly once from HBM
// via the Tensor Data Mover (double-buffered, overlapped with compute);
// all GEMMs use V_WMMA_F32_16X16X4_F32; the final 32x1024 accumulator
// lives in VGPRs (4 C-tiles / wave) across the whole d-loop.
// ---------------------------------------------------------------------------

typedef __attribute__((ext_vector_type(2))) float         v2f;
typedef __attribute__((ext_vector_type(8))) float         v8f;
typedef __attribute__((ext_vector_type(4))) unsigned int  v4u;
typedef __attribute__((ext_vector_type(8))) int           v8i_t;
typedef __attribute__((ext_vector_type(4))) int           v4i_t;

#define WMMA_F32(a, b, c) \
    __builtin_amdgcn_wmma_f32_16x16x4_f32(false, (a), false, (b), (short)0, (c), false, false)

// --- TDM: async 2-D tile load (64x64 f32 plane) global -> LDS --------------
__device__ __forceinline__ void tdm_load_plane_64x64(const float* gsrc, const float* ldst) {
    unsigned long long ga = (unsigned long long)gsrc;
    unsigned int lds_off  = (unsigned int)(unsigned long long)ldst;  // low 32b = LDS offset

    v4u g0;
    g0.x = 1u;                                       // count=1 (user D#), gather off
    g0.y = lds_off;                                  // lds_addr [63:32]
    g0.z = (unsigned int)(ga & 0xFFFFFFFFull);       // global_addr[31:0]
    g0.w = (unsigned int)((ga >> 32) & 0x01FFFFFFull)// global_addr[56:32]
         | (2u << 30);                               // type=2 ("image")

    v8i_t g1;
    g1[0] = (int)(2u << 16);      // data_size=2 -> 4 bytes; no multicast/pad/iterate
    g1[1] = (int)(64u << 16);     // tensor_dim0[15:0] = 64
    g1[2] = (int)(64u << 16);     // tensor_dim0 hi = 0, tensor_dim1[15:0] = 64
    g1[3] = (int)(64u << 16);     // tensor_dim1 hi = 0, tile_dim0 = 64
    g1[4] = (int)64;              // tile_dim1 = 64, tile_dim2 = 0
    g1[5] = (int)64;              // tensor_dim0_stride[31:0] = 64 (elements)
    g1[6] = 0;                    // stride hi, tensor_dim1_stride lo (unused, 2-D)
    g1[7] = 0;

    v4i_t gz4 = {0, 0, 0, 0};     // groups 2/3 unused (2-D tensor)
    v8i_t gz8 = {0, 0, 0, 0, 0, 0, 0, 0};
    __builtin_amdgcn_tensor_load_to_lds(g0, g1, gz4, gz4, gz8, 0);
}

__launch_bounds__(1024)
__global__ void spectral_pool_dct3d(const float* __restrict__ x, float* __restrict__ out) {
    __shared__ float lM[32 * 64];           //   8 KB : combined DCT matrix M[k][d]
    __shared__ float lX[2 * 4 * 64 * 64];   // 128 KB : double-buffered x planes
    __shared__ float lP[4 * 32 * 64];       //  32 KB : P[d][j][w] = M @ x[d]
    __shared__ float lQ[4 * 32 * 32];       //  16 KB : Q[d][j][i] = P[d] @ M^T

    const int tid  = threadIdx.x;
    const int wave = tid >> 5;
    const int lane = tid & 31;
    const int half = lane >> 4;          // 0: lanes 0-15, 1: lanes 16-31
    const int l16  = lane & 15;
    const int kb   = half << 1;          // K sub-offset for A/B fragments (0 or 2)
    const int bc   = blockIdx.x;         // 0..127 == (b*32 + c)

    const float* xslab = x + (size_t)bc * 64 * 64 * 64;

    // ---- Prefetch chunk 0 via TDM; overlaps with the M-matrix build --------
    if (wave == 0) {
        #pragma unroll
        for (int dl = 0; dl < 4; ++dl)
            tdm_load_plane_64x64(xslab + (size_t)dl * 4096, &lX[dl * 4096]);
    }

    // ---- Build M[k][d] = sum_{t<32} C32[t,k] * C64[t,d] --------------------
    for (int e = tid; e < 32 * 64; e += 1024) {
        const int k = e >> 6, d = e & 63;
        float acc = 0.f;
        const float a32 = 3.14159265358979f * (2.f * k + 1.f) / 64.f;
        const float a64 = 3.14159265358979f * (2.f * d + 1.f) / 128.f;
        for (int t = 0; t < 32; ++t) {
            const float s32 = (t == 0) ? 0.17677669529664f : 0.25f;            // sqrt(1/32), sqrt(2/32)
            const float s64 = (t == 0) ? 0.125f            : 0.17677669529664f; // sqrt(1/64), sqrt(2/64)
            acc += (s32 * __cosf(a32 * t)) * (s64 * __cosf(a64 * t));
        }
        lM[e] = acc;
    }
    __syncthreads();

    // ---- Persistent accumulator: 32 x 1024 (k x (j*32+i)), 4 tiles/wave ----
    const int kt  = wave >> 4;           // k-tile: 0 or 1
    const int ntb = (wave & 15) << 2;    // first of 4 N-tiles owned by this wave
    v8f acc[4] = {v8f{}, v8f{}, v8f{}, v8f{}};

    for (int c = 0; c < 16; ++c) {
        const int d0  = c << 2;
        const int buf = (c & 1) * 4 * 4096;

        // ---- Chunk c resident? Then kick off chunk c+1 into the other buf -
        if (wave == 0)
            __builtin_amdgcn_s_wait_tensorcnt(0);
        __syncthreads();
        if (wave == 0 && c + 1 < 16) {
            const int nb = ((c + 1) & 1) * 4 * 4096;
            #pragma unroll
            for (int dl = 0; dl < 4; ++dl)
                tdm_load_plane_64x64(xslab + (size_t)(d0 + 4 + dl) * 4096, &lX[nb + dl * 4096]);
        }

        // ---- Phase P: P[pl] = M(32x64) @ x[pl](64x64); 1 tile / wave ------
        {
            const int pl = wave >> 3, t = wave & 7, mt = t >> 2, nt = t & 3;
            const float* Mrow = &lM[(mt * 16 + l16) * 64];
            const float* Xp   = &lX[buf + pl * 4096 + nt * 16 + l16];
            v8f cc = {};
            #pragma unroll
            for (int k0 = 0; k0 < 64; k0 += 4) {
                v2f a, b;
                a.x = Mrow[k0 + kb];
                a.y = Mrow[k0 + kb + 1];
                b.x = Xp[(k0 + kb) * 64];
                b.y = Xp[(k0 + kb + 1) * 64];
                cc = WMMA_F32(a, b, cc);
            }
            float* Pp = &lP[pl * 2048 + (mt * 16 + half * 8) * 64 + nt * 16 + l16];
            #pragma unroll
            for (int v = 0; v < 8; ++v) Pp[v * 64] = cc[v];
        }
        __syncthreads();

        // ---- Phase Q: Q[pl] = P[pl](32x64) @ M^T(64x32); waves 0-15 -------
        if (wave < 16) {
            const int pl = wave >> 2, mt = (wave >> 1) & 1, nt = wave & 1;
            const float* Prow = &lP[pl * 2048 + (mt * 16 + l16) * 64];
            const float* Mcol = &lM[(nt * 16 + l16) * 64];  // B[K=w,N=i] = M[i][w]
            v8f cc = {};
            #pragma unroll
            for (int k0 = 0; k0 < 64; k0 += 4) {
                v2f a, b;
                a.x = Prow[k0 + kb];
                a.y = Prow[k0 + kb + 1];
                b.x = Mcol[k0 + kb];
                b.y = Mcol[k0 + kb + 1];
                cc = WMMA_F32(a, b, cc);
            }
            float* Qp = &lQ[pl * 1024 + (mt * 16 + half * 8) * 32 + nt * 16 + l16];
            #pragma unroll
            for (int v = 0; v < 8; ++v) Qp[v * 32] = cc[v];
        }
        __syncthreads();

        // ---- Phase D: acc += M[:, d0:d0+4] (32x4) @ Q (4x1024) ------------
        {
            const float* Mr = &lM[(kt * 16 + l16) * 64 + d0 + kb];
            v2f a;
            a.x = Mr[0];
            a.y = Mr[1];
            #pragma unroll
            for (int t = 0; t < 4; ++t) {
                const int N = (ntb + t) * 16 + l16;
                v2f b;
                b.x = lQ[(kb + 0) * 1024 + N];
                b.y = lQ[(kb + 1) * 1024 + N];
                acc[t] = WMMA_F32(a, b, acc[t]);
            }
        }
        __syncthreads();   // lQ consumed; safe to overwrite lP/lQ next iter
    }

    // ---- Store: out[bc][k][j][i], N = j*32+i --------------------------------
    float* outp = out + (size_t)bc * 32 * 32 * 32;
    #pragma unroll
    for (int t = 0; t < 4; ++t) {
        const int N = (ntb + t) * 16 + l16;
        #pragma unroll
        for (int v = 0; v < 8; ++v) {
            const int k = kt * 16 + v + half * 8;
            outp[k * 1024 + N] = acc[t][v];
        }
    }
}

extern "C" void kernel_launch(void* const* d_in, const int* in_sizes, int n_in,
                              void* d_out, int out_size, void* d_ws, size_t ws_size,
                              hipStream_t stream) {
    (void)in_sizes; (void)n_in; (void)d_ws; (void)ws_size; (void)out_size;
    const float* x = (const float*)d_in[0];   // (4,32,64,64,64) f32
    float* out     = (float*)d_out;           // (4,32,32,32,32) f32
    spectral_pool_dct3d<<<dim3(128), dim3(1024), 0, stream>>>(x, out);
}